// CapsuleLayer_80779744903485
// MI455X (gfx1250) — compile-verified
//
#include <hip/hip_runtime.h>
#include <math.h>

// CDNA5 / gfx1250. wave32. f32 WMMA 16x16x4.
typedef __attribute__((ext_vector_type(2))) float v2f;
typedef __attribute__((ext_vector_type(8))) float v8f;

#define CAPS_B    64
#define CAPS_NI   2048
#define CAPS_NJ   32
#define CAPS_DOUT 32
#define CAPS_DK   16
#define CAPS_JD   1024            // NJ * DOUT
#define ICHUNK    32              // i's per workgroup
#define NBLK      256             // 4 b-tiles x (2048/ICHUNK) i-chunks

// LDS padded layout: jd-row (b) stride 1058 (= 33*32+2, 8*1058 % 64 == 16 -> the two
// lane-halves of a tile store hit disjoint bank sets); j stride 33 (lane*33 mod 64 is
// injective over 32 lanes -> conflict-free per-lane d-loops in the logit phase).
#define PJ 33
#define PB (PJ * CAPS_NJ + 2)     // 1058 floats per local-b row

// One 16(b) x 16(jd) u_hat tile, K=16 via four K=4 f32 WMMA steps.
// A layout (16x4 f32): lanes0-15 hold M=lane, K = ks+{0,1}; lanes16-31 M=lane-16, K = ks+{2,3}.
// B layout (4x16 f32): mirrored, N across lanes, same K-per-lane-half pairing.
__device__ __forceinline__ v8f uhat_tile_k16(const v2f a[4], const float* __restrict__ wp)
{
  v8f c;
#pragma unroll
  for (int e = 0; e < 8; ++e) c[e] = 0.0f;
#pragma unroll
  for (int st = 0; st < 4; ++st) {
    v2f bv = *(const v2f*)(wp + 4 * st);
    c = __builtin_amdgcn_wmma_f32_16x16x4_f32(false, a[st], false, bv, (short)0, c,
                                              false, false);
  }
  return c;
}

// One fused routing pass over all (b,i):
//   routed==0 : s += (1/32) * u_hat                        (softmax of zero logits)
//   routed==1 : logits[b,i,j] = sum_d u_hat * vin[b,j,d]; c = softmax_j; s += c * u_hat
// u_hat is recomputed per pass (never hits HBM); within a pass each tile is computed by
// WMMA exactly once, held in VGPRs for the weighted accumulate, and mirrored to LDS only
// for the per-(b,i,j) logit reduction. s reduced across i-chunks with global f32 atomics.
__global__ __launch_bounds__(256)
void caps_pass_kernel(const float* __restrict__ x, const float* __restrict__ W,
                      const float* __restrict__ vin, float* __restrict__ s_out,
                      int routed)
{
  __shared__ float ush[16 * PB];         // u_hat[b_local][j][d], padded
  __shared__ float vsh[16 * PB];         // vin  [b_local][j][d], padded
  __shared__ float blog[16 * CAPS_NJ];   // routing logits
  __shared__ float csm [16 * CAPS_NJ];   // softmax(blog) over j

  const int tid  = threadIdx.x;
  const int wave = tid >> 5;
  const int lane = tid & 31;
  const int lh   = lane >> 4;   // lane half
  const int ln   = lane & 15;

  const int bt = blockIdx.x & 3;           // which 16-batch tile
  const int b0 = bt << 4;
  const int i0 = (blockIdx.x >> 2) * ICHUNK;

  // stage v for this b-tile into LDS once (invariant over the i-loop)
  if (routed) {
    for (int idx = tid; idx < 16 * CAPS_JD; idx += 256) {
      const int b = idx >> 10, jd = idx & 1023;
      vsh[b * PB + (jd >> 5) * PJ + (jd & 31)] = vin[(size_t)(b0 + b) * CAPS_JD + jd];
    }
    __syncthreads();
  }

  // per-wave s accumulators: tile t covers jd-tile n = wave + 8t
  v8f sacc[8];
#pragma unroll
  for (int t = 0; t < 8; ++t)
#pragma unroll
    for (int e = 0; e < 8; ++e) sacc[t][e] = 0.0f;

  for (int ii = 0; ii < ICHUNK; ++ii) {
    const int i = i0 + ii;

    // A operand: x[b0+ln, i, :]  (shared by every jd tile of this wave)
    const float* xp = x + ((size_t)(b0 + ln) * CAPS_NI + i) * CAPS_DK + 2 * lh;
    v2f a[4];
#pragma unroll
    for (int st = 0; st < 4; ++st) a[st] = *(const v2f*)(xp + 4 * st);

    const float* wbase = W + (size_t)i * (CAPS_JD * CAPS_DK);

    if (!routed) {
      // iteration 0: uniform coupling 1/32
#pragma unroll
      for (int t = 0; t < 8; ++t) {
        const int n = wave + (t << 3);
        const float* wp = wbase + (size_t)(n * 16 + ln) * CAPS_DK + 2 * lh;
        v8f c = uhat_tile_k16(a, wp);
#pragma unroll
        for (int r = 0; r < 8; ++r) sacc[t][r] += c[r] * (1.0f / 32.0f);
      }
    } else {
      // phase A: WMMA tiles -> VGPRs (kept) + LDS mirror (for the logit reduction)
      v8f ut[8];
#pragma unroll
      for (int t = 0; t < 8; ++t) {
        const int n = wave + (t << 3);
        const float* wp = wbase + (size_t)(n * 16 + ln) * CAPS_DK + 2 * lh;
        ut[t] = uhat_tile_k16(a, wp);
        const int ubase = (n >> 1) * PJ + (n & 1) * 16 + ln;  // j*PJ + d
#pragma unroll
        for (int r = 0; r < 8; ++r)
          ush[(r + (lh << 3)) * PB + ubase] = ut[t][r];
      }
      __syncthreads();

      // phase B: logits. wave w owns b = w, w+8; lane = j; per-lane d-loop
      // (conflict-free LDS, no shuffles, no atomics)
#pragma unroll
      for (int h = 0; h < 2; ++h) {
        const int b = wave + (h << 3);
        const float* up = &ush[b * PB + lane * PJ];
        const float* vp = &vsh[b * PB + lane * PJ];
        float acc = 0.0f;
#pragma unroll
        for (int d = 0; d < 32; ++d) acc += up[d] * vp[d];
        blog[b * CAPS_NJ + lane] = acc;
      }
      __syncthreads();

      // softmax over the 32 output capsules, one thread per local b
      if (tid < 16) {
        float mx = -3.0e38f;
        for (int j = 0; j < CAPS_NJ; ++j) mx = fmaxf(mx, blog[tid * CAPS_NJ + j]);
        float sum = 0.0f;
        for (int j = 0; j < CAPS_NJ; ++j) {
          float e = __expf(blog[tid * CAPS_NJ + j] - mx);
          csm[tid * CAPS_NJ + j] = e;
          sum += e;
        }
        const float inv = 1.0f / sum;
        for (int j = 0; j < CAPS_NJ; ++j) csm[tid * CAPS_NJ + j] *= inv;
      }
      __syncthreads();

      // phase C: s += c_route * u_hat, straight from the VGPR copy
#pragma unroll
      for (int t = 0; t < 8; ++t) {
        const int j = (wave + (t << 3)) >> 1;
#pragma unroll
        for (int r = 0; r < 8; ++r) {
          const float cr = csm[(r + (lh << 3)) * CAPS_NJ + j];  // LDS broadcast
          sacc[t][r] += cr * ut[t][r];
        }
      }
    }
  }

  // flush partial s[b,jd] (reduction over i-chunks) with global f32 atomics
#pragma unroll
  for (int t = 0; t < 8; ++t) {
    const int n = wave + (t << 3);
#pragma unroll
    for (int r = 0; r < 8; ++r) {
      const int bl = r + (lh << 3);
      atomicAdd(&s_out[(size_t)(b0 + bl) * CAPS_JD + n * 16 + ln], sacc[t][r]);
    }
  }
}

__global__ __launch_bounds__(256)
void zero_kernel(float* __restrict__ p, int n)
{
  const int idx = blockIdx.x * 256 + threadIdx.x;
  if (idx < n) p[idx] = 0.0f;
}

// v = squash(s) per (b,j) row of 32 d's; optionally out = squash(s) + addv  (v-history sum)
__global__ __launch_bounds__(256)
void squash_kernel(const float* __restrict__ s, const float* __restrict__ addv,
                   float* __restrict__ out)
{
  const int row  = blockIdx.x * 8 + (threadIdx.x >> 5);  // b*32+j, 2048 rows
  const int lane = threadIdx.x & 31;                      // d
  const int idx  = row * 32 + lane;
  const float val = s[idx];
  float sq = val * val;
  sq += __shfl_xor(sq, 1, 32);
  sq += __shfl_xor(sq, 2, 32);
  sq += __shfl_xor(sq, 4, 32);
  sq += __shfl_xor(sq, 8, 32);
  sq += __shfl_xor(sq, 16, 32);
  const float eps = 1e-9f;
  float v = val * (sq / (1.0f + sq + eps)) / sqrtf(sq + eps);
  if (addv) v += addv[idx];
  out[idx] = v;
}

extern "C" void kernel_launch(void* const* d_in, const int* in_sizes, int n_in,
                              void* d_out, int out_size, void* d_ws, size_t ws_size,
                              hipStream_t stream)
{
  (void)in_sizes; (void)n_in; (void)out_size; (void)ws_size;
  const float* x = (const float*)d_in[0];
  const float* W = (const float*)d_in[1];   // [1,2048,32,32,16], k contiguous
  float* out  = (float*)d_out;              // [64,32,32]
  float* s    = (float*)d_ws;               // 256 KB accumulator
  float* v0   = s + 65536;                  // v after iter 0
  float* vsum = s + 2 * 65536;              // v0 + v1 (logits are linear in v history)

  const dim3 blk(256), gpass(NBLK), gsmall(256);

  // routing iteration 0: c uniform (softmax of zero logits)
  zero_kernel   <<<gsmall, blk, 0, stream>>>(s, 65536);
  caps_pass_kernel<<<gpass, blk, 0, stream>>>(x, W, nullptr, s, 0);
  squash_kernel <<<gsmall, blk, 0, stream>>>(s, nullptr, v0);

  // routing iteration 1: logits from v0
  zero_kernel   <<<gsmall, blk, 0, stream>>>(s, 65536);
  caps_pass_kernel<<<gpass, blk, 0, stream>>>(x, W, v0, s, 1);
  squash_kernel <<<gsmall, blk, 0, stream>>>(s, v0, vsum);   // vsum = v0 + v1

  // routing iteration 2: logits from v0+v1  (b2 = sum_d u_hat*(v0+v1))
  zero_kernel   <<<gsmall, blk, 0, stream>>>(s, 65536);
  caps_pass_kernel<<<gpass, blk, 0, stream>>>(x, W, vsum, s, 1);
  squash_kernel <<<gsmall, blk, 0, stream>>>(s, nullptr, out);
}